// DGCNN_57037165691323
// MI455X (gfx1250) — compile-verified
//
#include <hip/hip_runtime.h>
#include <hip/hip_bf16.h>

// ---------------------------------------------------------------------------
// DGCNN forward for MI455X (gfx1250, wave32). All heavy GEMMs use the fp32
// CDNA5 WMMA op V_WMMA_F32_16X16X4_F32. WMMA inner loops are branch-free:
// operands come from zero-padded LDS tiles / padded-transposed weight panels.
// LDS panel strides are chosen bank-conflict-free for wave32 on 64 banks.
// ---------------------------------------------------------------------------

typedef float v2f __attribute__((ext_vector_type(2)));
typedef float v8f __attribute__((ext_vector_type(8)));

static constexpr int BB  = 8;      // batch
static constexpr int PN  = 1024;   // points
static constexpr int KNN = 20;     // neighbors
static constexpr int NKC = PN * KNN;
static constexpr float EPS = 1e-5f;
static constexpr int XMS = 72;     // col-panel row stride (words): 2*XMS % 64 == 16
                                   // -> half-waves hit disjoint bank groups

__device__ __forceinline__ float lrelu_f(float v) { return v >= 0.f ? v : 0.2f * v; }

__device__ __forceinline__ v8f wmma4(v2f a, v2f b, v8f c) {
  // D = A(16x4, f32) * B(4x16, f32) + C(16x16, f32)
  return __builtin_amdgcn_wmma_f32_16x16x4_f32(false, a, false, b, (short)0, c,
                                               false, false);
}

// ---------------------------------------------------------------------------
// 0) transpose features (B,N,3) -> x0 (B,3,N)
// ---------------------------------------------------------------------------
__global__ __launch_bounds__(256) void dgcnn_transpose_kernel(
    const float* __restrict__ feats, float* __restrict__ x0) {
  int t = blockIdx.x * 256 + threadIdx.x;
  if (t >= BB * PN) return;
  int b = t / PN, n = t - b * PN;
  for (int c = 0; c < 3; ++c)
    x0[((size_t)b * 3 + c) * PN + n] = feats[((size_t)b * PN + n) * 3 + c];
}

// ---------------------------------------------------------------------------
// 0b) weight prep: WT[c][o] = c < CK ? W[o][c] : 0   (c < CKp, CKp mult of 4)
//     Transposed so WMMA A-fragment loads are lane-coalesced; zero-padded so
//     the WMMA K-loop needs no guards.
// ---------------------------------------------------------------------------
__global__ __launch_bounds__(256) void dgcnn_padwT_kernel(
    const float* __restrict__ W, float* __restrict__ WT,
    int O, int ldw, int CK, int CKp) {
  int t = blockIdx.x * 256 + threadIdx.x;
  if (t >= CKp * O) return;
  int c = t / O, o = t - c * O;
  WT[t] = (c < CK) ? W[(size_t)o * ldw + c] : 0.f;
}

// ---------------------------------------------------------------------------
// 1) per-point squared norm, x layout (B,C,N) with batch stride bS
// ---------------------------------------------------------------------------
__global__ __launch_bounds__(256) void dgcnn_sq_kernel(
    const float* __restrict__ x, int bS, int C, float* __restrict__ sq) {
  int t = blockIdx.x * 256 + threadIdx.x;
  if (t >= BB * PN) return;
  int b = t / PN, n = t - b * PN;
  const float* xb = x + (size_t)b * bS;
  float s = 0.f;
  for (int c = 0; c < C; ++c) { float v = xb[(size_t)c * PN + n]; s += v * v; }
  sq[(size_t)b * PN + n] = s;
}

// ---------------------------------------------------------------------------
// 2) pairwise distances D[b][n][m] = |x_n|^2 + |x_m|^2 - 2 x_n.x_m via WMMA
//    4-wave block computes a 16x64 tile; both panels staged in LDS (padded),
//    WMMA loop is branch-free LDS-only. xm stride padded for bank freedom.
// ---------------------------------------------------------------------------
__global__ __launch_bounds__(128) void dgcnn_dist_kernel(
    const float* __restrict__ x, const float* __restrict__ sq,
    float* __restrict__ D, int C, int bS) {
  __shared__ float xa[128 * 16];    // row panel  (Cp x 16)
  __shared__ float xm[128 * XMS];   // col panel  (Cp x 64, stride 72)
  int b = blockIdx.z;
  int n0 = blockIdx.x * 16;
  int m0 = blockIdx.y * 64;
  int tid = threadIdx.x;
  int Cp = (C + 3) & ~3;
  const float* xb = x + (size_t)b * bS;
  // stage row panel: xa[c*16+col] = x[c][n0+col]
  for (int ev = tid; ev < Cp * 4; ev += 128) {
    int c = ev >> 2, colv = (ev & 3) * 4;
    float4 v = make_float4(0.f, 0.f, 0.f, 0.f);
    if (c < C) v = *(const float4*)(xb + (size_t)c * PN + n0 + colv);
    *(float4*)(xa + c * 16 + colv) = v;
  }
  // stage col panel: xm[c*XMS+col] = x[c][m0+col]
  for (int ev = tid; ev < Cp * 16; ev += 128) {
    int c = ev >> 4, colv = (ev & 15) * 4;
    float4 v = make_float4(0.f, 0.f, 0.f, 0.f);
    if (c < C) v = *(const float4*)(xb + (size_t)c * PN + m0 + colv);
    *(float4*)(xm + c * XMS + colv) = v;
  }
  __syncthreads();
  int wave = tid >> 5;
  int l = tid & 31;
  int row = l & 15, half = l >> 4;
  const float* xmw = xm + wave * 16;
  v8f acc = {};
#pragma unroll 4
  for (int kk = 0; kk < Cp; kk += 4) {
    int c0 = kk + 2 * half;
    v2f a, bf;
    a.x = xa[c0 * 16 + row];
    a.y = xa[(c0 + 1) * 16 + row];
    bf.x = xmw[c0 * XMS + row];
    bf.y = xmw[(c0 + 1) * XMS + row];
    acc = wmma4(a, bf, acc);
  }
  const float* sqb = sq + (size_t)b * PN;
  int mcol = m0 + wave * 16 + row;
  float sqm = sqb[mcol];
  float* Db = D + (size_t)b * PN * PN;
#pragma unroll
  for (int r = 0; r < 8; ++r) {
    int rr = n0 + r + 8 * half;
    Db[(size_t)rr * PN + mcol] = sqb[rr] + sqm - 2.0f * acc[r];
  }
}

// ---------------------------------------------------------------------------
// 3) top-K smallest per row (ties -> lower index, matching jax.lax.top_k)
// ---------------------------------------------------------------------------
__global__ __launch_bounds__(256) void dgcnn_topk_kernel(
    const float* __restrict__ D, int* __restrict__ idx) {
  int rowid = blockIdx.x;               // b*PN + n
  __shared__ float dist[PN];
  __shared__ float rv[256];
  __shared__ int   ri[256];
  int tid = threadIdx.x;
  const float* drow = D + (size_t)rowid * PN;
  for (int i = tid; i < PN; i += 256) dist[i] = drow[i];
  __syncthreads();
  for (int kk = 0; kk < KNN; ++kk) {
    float bv = 3.4e38f; int bi = PN;
    for (int i = tid; i < PN; i += 256) {
      float v = dist[i];
      if (v < bv || (v == bv && i < bi)) { bv = v; bi = i; }
    }
    rv[tid] = bv; ri[tid] = bi;
    __syncthreads();
    for (int s = 128; s > 0; s >>= 1) {
      if (tid < s) {
        float v2 = rv[tid + s]; int i2 = ri[tid + s];
        if (v2 < rv[tid] || (v2 == rv[tid] && i2 < ri[tid])) { rv[tid] = v2; ri[tid] = i2; }
      }
      __syncthreads();
    }
    if (tid == 0) {
      idx[(size_t)rowid * KNN + kk] = ri[0];
      dist[ri[0]] = 3.4e38f;
    }
    __syncthreads();
  }
}

// ---------------------------------------------------------------------------
// 4) edge-conv GEMM with fused neighbor gather:
//    tmp[b][o][j] = sum_c WT[c][o] * feat[c][j],  j = n*K + k
//    feat[c][j]   = c<C ? x[c][idx]-x[c][n] : x[c-C][n]   (zero-padded to CKp)
//    A loads are lane-coalesced from the padded-transposed weight panel; the
//    WMMA loop is branch-free.
// ---------------------------------------------------------------------------
__global__ __launch_bounds__(128) void dgcnn_edgeconv_kernel(
    const float* __restrict__ WT, int CKp,
    const float* __restrict__ x, int bS, int C, int CK,
    const int* __restrict__ idx, float* __restrict__ tmp, int O) {
  __shared__ float feat[256 * 16];     // CKp <= 256
  __shared__ int nl[16], ml[16];
  int b = blockIdx.z;
  int j0 = blockIdx.x * 16;
  int tid = threadIdx.x;
  if (tid < 16) {
    int j = j0 + tid;
    int n = j / KNN, k = j - n * KNN;
    nl[tid] = n;
    ml[tid] = idx[((size_t)b * PN + n) * KNN + k];
  }
  __syncthreads();
  const float* xb = x + (size_t)b * bS;
  for (int e = tid; e < CKp * 16; e += 128) {
    int c = e >> 4, col = e & 15;
    float v = 0.f;
    if (c < CK) {
      if (c < C) v = xb[(size_t)c * PN + ml[col]] - xb[(size_t)c * PN + nl[col]];
      else       v = xb[(size_t)(c - C) * PN + nl[col]];
    }
    feat[e] = v;
  }
  __syncthreads();
  int wave = tid >> 5;
  int l = tid & 31;
  int row = l & 15, half = l >> 4;
  int o0 = (blockIdx.y * 4 + wave) * 16;
  const float* Wc = WT + o0 + row;     // column o0+row of W (padded, transposed)
  v8f acc = {};
#pragma unroll 4
  for (int kk = 0; kk < CKp; kk += 4) {
    int c0 = kk + 2 * half;
    v2f a, bf;
    a.x = Wc[(size_t)c0 * O];
    a.y = Wc[(size_t)(c0 + 1) * O];
    bf.x = feat[c0 * 16 + row];
    bf.y = feat[(c0 + 1) * 16 + row];
    acc = wmma4(a, bf, acc);
  }
#pragma unroll
  for (int r = 0; r < 8; ++r) {
    int o = o0 + r + 8 * half;
    tmp[((size_t)b * O + o) * NKC + j0 + row] = acc[r];
  }
}

// ---------------------------------------------------------------------------
// 5) per-channel batch statistics over (B, M): mean + 1/sqrt(var+eps)
// ---------------------------------------------------------------------------
__global__ __launch_bounds__(256) void dgcnn_bnstats_kernel(
    const float* __restrict__ t, float* __restrict__ mean,
    float* __restrict__ invstd, int O, int M) {
  int o = blockIdx.x;
  __shared__ float s1[256], s2[256];
  int tid = threadIdx.x;
  float a = 0.f, q = 0.f;
  for (int b = 0; b < BB; ++b) {
    const float* p = t + ((size_t)b * O + o) * M;
    for (int i = tid; i < M; i += 256) { float v = p[i]; a += v; q += v * v; }
  }
  s1[tid] = a; s2[tid] = q;
  __syncthreads();
  for (int s = 128; s > 0; s >>= 1) {
    if (tid < s) { s1[tid] += s1[tid + s]; s2[tid] += s2[tid + s]; }
    __syncthreads();
  }
  if (tid == 0) {
    float inv = 1.f / (float)(BB * M);
    float m = s1[0] * inv;
    float var = s2[0] * inv - m * m;
    mean[o] = m;
    invstd[o] = rsqrtf(var + EPS);
  }
}

// ---------------------------------------------------------------------------
// 6) BN + LeakyReLU + max over K, writes into a channel slice of xc (B,512,N)
// ---------------------------------------------------------------------------
__global__ __launch_bounds__(256) void dgcnn_bnactmax_kernel(
    const float* __restrict__ tmp, const float* __restrict__ mean,
    const float* __restrict__ invstd, const float* __restrict__ g,
    const float* __restrict__ bb, float* __restrict__ xc, int O, int choff) {
  size_t t = (size_t)blockIdx.x * 256 + threadIdx.x;
  if (t >= (size_t)BB * O * PN) return;
  int n = (int)(t % PN);
  int o = (int)((t / PN) % O);
  int b = (int)(t / ((size_t)PN * O));
  float sc = invstd[o] * g[o];
  float sh = bb[o] - mean[o] * sc;
  const float* p = tmp + ((size_t)b * O + o) * NKC + (size_t)n * KNN;
  float mx = -3.4e38f;
  for (int k = 0; k < KNN; ++k) {
    float v = lrelu_f(p[k] * sc + sh);
    mx = fmaxf(mx, v);
  }
  xc[((size_t)b * 512 + choff + o) * PN + n] = mx;
}

// ---------------------------------------------------------------------------
// 7) conv5: x5[b][o][n] = sum_c W5T[c][o] * xc[b][c][n], O=1024, C=512 (WMMA)
// ---------------------------------------------------------------------------
__global__ __launch_bounds__(128) void dgcnn_conv5_kernel(
    const float* __restrict__ W5T, const float* __restrict__ xc,
    float* __restrict__ tmp5) {
  __shared__ float xcs[512 * 16];      // 32 KB
  int b = blockIdx.z;
  int n0 = blockIdx.x * 16;
  int tid = threadIdx.x;
  const float* xcb = xc + (size_t)b * 512 * PN;
  for (int ev = tid; ev < 512 * 4; ev += 128) {
    int c = ev >> 2, colv = (ev & 3) * 4;
    *(float4*)(xcs + c * 16 + colv) =
        *(const float4*)(xcb + (size_t)c * PN + n0 + colv);
  }
  __syncthreads();
  int wave = tid >> 5;
  int l = tid & 31;
  int row = l & 15, half = l >> 4;
  int o0 = (blockIdx.y * 4 + wave) * 16;
  const float* Wc = W5T + o0 + row;
  v8f acc = {};
#pragma unroll 4
  for (int kk = 0; kk < 512; kk += 4) {
    int c0 = kk + 2 * half;
    v2f a, bf;
    a.x = Wc[(size_t)c0 * 1024];
    a.y = Wc[(size_t)(c0 + 1) * 1024];
    bf.x = xcs[c0 * 16 + row];
    bf.y = xcs[(c0 + 1) * 16 + row];
    acc = wmma4(a, bf, acc);
  }
#pragma unroll
  for (int r = 0; r < 8; ++r) {
    int o = o0 + r + 8 * half;
    tmp5[((size_t)b * 1024 + o) * PN + n0 + row] = acc[r];
  }
}

// ---------------------------------------------------------------------------
// 8) BN + LeakyReLU in place on tmp5 (B,1024,N)
// ---------------------------------------------------------------------------
__global__ __launch_bounds__(256) void dgcnn_bnact5_kernel(
    float* __restrict__ tmp5, const float* __restrict__ mean,
    const float* __restrict__ invstd, const float* __restrict__ g,
    const float* __restrict__ bb) {
  size_t t = (size_t)blockIdx.x * 256 + threadIdx.x;
  if (t >= (size_t)BB * 1024 * PN) return;
  int o = (int)((t / PN) & 1023);
  float sc = invstd[o] * g[o];
  float sh = bb[o] - mean[o] * sc;
  tmp5[t] = lrelu_f(tmp5[t] * sc + sh);
}

// ---------------------------------------------------------------------------
// 9) global pooling: pooled[b][0:1024]=max_n, pooled[b][1024:2048]=mean_n
// ---------------------------------------------------------------------------
__global__ __launch_bounds__(256) void dgcnn_pool_kernel(
    const float* __restrict__ x5, float* __restrict__ pooled) {
  int bo = blockIdx.x;                 // b*1024 + o
  __shared__ float smx[256], ssm[256];
  int tid = threadIdx.x;
  const float* p = x5 + (size_t)bo * PN;
  float mx = -3.4e38f, sm = 0.f;
  for (int i = tid; i < PN; i += 256) { float v = p[i]; mx = fmaxf(mx, v); sm += v; }
  smx[tid] = mx; ssm[tid] = sm;
  __syncthreads();
  for (int s = 128; s > 0; s >>= 1) {
    if (tid < s) { smx[tid] = fmaxf(smx[tid], smx[tid + s]); ssm[tid] += ssm[tid + s]; }
    __syncthreads();
  }
  if (tid == 0) {
    int b = bo >> 10, o = bo & 1023;
    pooled[(size_t)b * 2048 + o] = smx[0];
    pooled[(size_t)b * 2048 + 1024 + o] = ssm[0] * (1.f / PN);
  }
}

// ---------------------------------------------------------------------------
// 10) FC1 h = pooled @ Wl1^T  (8x2048 . 2048x512)
// ---------------------------------------------------------------------------
__global__ __launch_bounds__(128) void dgcnn_fc1_kernel(
    const float* __restrict__ pooled, const float* __restrict__ Wl1,
    float* __restrict__ h) {
  int t = blockIdx.x * 128 + threadIdx.x;
  if (t >= BB * 512) return;
  int b = t / 512, j = t - b * 512;
  const float* p = pooled + (size_t)b * 2048;
  const float* w = Wl1 + (size_t)j * 2048;
  float s = 0.f;
  for (int c = 0; c < 2048; ++c) s += p[c] * w[c];
  h[t] = s;
}

// 11) BN over batch (8 values / channel) + LeakyReLU
__global__ __launch_bounds__(512) void dgcnn_fc1bn_kernel(
    const float* __restrict__ h, const float* __restrict__ g6,
    const float* __restrict__ b6, float* __restrict__ hact) {
  int j = threadIdx.x;
  float a = 0.f, q = 0.f;
  for (int b = 0; b < BB; ++b) { float v = h[(size_t)b * 512 + j]; a += v; q += v * v; }
  float m = a * (1.f / BB);
  float var = q * (1.f / BB) - m * m;
  float inv = rsqrtf(var + EPS);
  float sc = inv * g6[j];
  float sh = b6[j] - m * sc;
  for (int b = 0; b < BB; ++b)
    hact[(size_t)b * 512 + j] = lrelu_f(h[(size_t)b * 512 + j] * sc + sh);
}

// 12) FC2 out = hact @ Wl2^T + bl2  (8x512 . 512x256)
__global__ __launch_bounds__(128) void dgcnn_fc2_kernel(
    const float* __restrict__ hact, const float* __restrict__ Wl2,
    const float* __restrict__ bl2, float* __restrict__ out) {
  int t = blockIdx.x * 128 + threadIdx.x;
  if (t >= BB * 256) return;
  int b = t / 256, j = t - b * 256;
  const float* p = hact + (size_t)b * 512;
  const float* w = Wl2 + (size_t)j * 512;
  float s = bl2[j];
  for (int c = 0; c < 512; ++c) s += p[c] * w[c];
  out[t] = s;
}

// ---------------------------------------------------------------------------
// Host orchestration
// ---------------------------------------------------------------------------
extern "C" void kernel_launch(void* const* d_in, const int* in_sizes, int n_in,
                              void* d_out, int out_size, void* d_ws, size_t ws_size,
                              hipStream_t stream) {
  (void)in_sizes; (void)n_in; (void)out_size; (void)ws_size;
  const float* feats = (const float*)d_in[1];
  const float* Wc4[4] = {(const float*)d_in[2], (const float*)d_in[5],
                         (const float*)d_in[8], (const float*)d_in[11]};
  const float* gc[4] = {(const float*)d_in[3], (const float*)d_in[6],
                        (const float*)d_in[9], (const float*)d_in[12]};
  const float* bc[4] = {(const float*)d_in[4], (const float*)d_in[7],
                        (const float*)d_in[10], (const float*)d_in[13]};
  const float* W5  = (const float*)d_in[14];
  const float* g5  = (const float*)d_in[15];
  const float* b5  = (const float*)d_in[16];
  const float* Wl1 = (const float*)d_in[17];
  const float* g6  = (const float*)d_in[18];
  const float* b6  = (const float*)d_in[19];
  const float* Wl2 = (const float*)d_in[20];
  const float* bl2 = (const float*)d_in[21];
  float* out = (float*)d_out;

  // workspace layout (floats)
  float* ws = (float*)d_ws;
  size_t off = 0;
  float* x0     = ws + off; off += (size_t)BB * 3 * PN;
  float* xc     = ws + off; off += (size_t)BB * 512 * PN;      // 4 M
  float* sqb    = ws + off; off += (size_t)BB * PN;
  float* Dm     = ws + off; off += (size_t)BB * PN * PN;       // 8 M
  int*   idxb   = (int*)(ws + off); off += (size_t)BB * PN * KNN;
  float* tmpb   = ws + off; off += (size_t)BB * 256 * NKC;     // 41.9 M (max layer)
  float* wpadT  = ws + off; off += (size_t)512 * 1024;         // padded/transposed W
  float* meanb  = ws + off; off += 1024;
  float* invb   = ws + off; off += 1024;
  float* pooled = ws + off; off += (size_t)BB * 2048;
  float* hbuf   = ws + off; off += (size_t)BB * 512;
  float* hact   = ws + off; off += (size_t)BB * 512;

  dgcnn_transpose_kernel<<<(BB * PN + 255) / 256, 256, 0, stream>>>(feats, x0);

  // Edge-conv layers. Layer 4: W4 is (256,512) while the layer-4 graph feature
  // has 2*128=256 channels; we use W4[:, :256] (ldw=512, CK=256) — the
  // dimensionally consistent DGCNN interpretation.
  const int Cin[4]   = {3, 64, 64, 128};
  const int CK[4]    = {6, 128, 128, 256};
  const int ldw[4]   = {6, 128, 128, 512};
  const int Oc[4]    = {64, 64, 128, 256};
  const int inoff[4] = {0, 0, 64, 128};
  const int outoff[4]= {0, 64, 128, 256};

  for (int li = 0; li < 4; ++li) {
    const float* xbase = (li == 0) ? x0 : (xc + (size_t)inoff[li] * PN);
    int bS = (li == 0) ? 3 * PN : 512 * PN;
    int C = Cin[li], O = Oc[li];
    int CKp = (CK[li] + 3) & ~3;

    dgcnn_padwT_kernel<<<(CKp * O + 255) / 256, 256, 0, stream>>>(
        Wc4[li], wpadT, O, ldw[li], CK[li], CKp);
    dgcnn_sq_kernel<<<(BB * PN + 255) / 256, 256, 0, stream>>>(xbase, bS, C, sqb);
    dgcnn_dist_kernel<<<dim3(PN / 16, PN / 64, BB), 128, 0, stream>>>(xbase, sqb, Dm, C, bS);
    dgcnn_topk_kernel<<<BB * PN, 256, 0, stream>>>(Dm, idxb);
    dgcnn_edgeconv_kernel<<<dim3(NKC / 16, O / 64, BB), 128, 0, stream>>>(
        wpadT, CKp, xbase, bS, C, CK[li], idxb, tmpb, O);
    dgcnn_bnstats_kernel<<<O, 256, 0, stream>>>(tmpb, meanb, invb, O, NKC);
    dgcnn_bnactmax_kernel<<<(int)(((size_t)BB * O * PN + 255) / 256), 256, 0, stream>>>(
        tmpb, meanb, invb, gc[li], bc[li], xc, O, outoff[li]);
  }

  // conv5 + BN + lrelu
  dgcnn_padwT_kernel<<<(512 * 1024 + 255) / 256, 256, 0, stream>>>(
      W5, wpadT, 1024, 512, 512, 512);
  dgcnn_conv5_kernel<<<dim3(PN / 16, 1024 / 64, BB), 128, 0, stream>>>(wpadT, xc, tmpb);
  dgcnn_bnstats_kernel<<<1024, 256, 0, stream>>>(tmpb, meanb, invb, 1024, PN);
  dgcnn_bnact5_kernel<<<(int)(((size_t)BB * 1024 * PN + 255) / 256), 256, 0, stream>>>(
      tmpb, meanb, invb, g5, b5);

  // pooling + FC head
  dgcnn_pool_kernel<<<BB * 1024, 256, 0, stream>>>(tmpb, pooled);
  dgcnn_fc1_kernel<<<(BB * 512 + 127) / 128, 128, 0, stream>>>(pooled, Wl1, hbuf);
  dgcnn_fc1bn_kernel<<<1, 512, 0, stream>>>(hbuf, g6, b6, hact);
  dgcnn_fc2_kernel<<<(BB * 256 + 127) / 128, 128, 0, stream>>>(hact, Wl2, bl2, out);
}